// ProjectSimplexModule_33011118637759
// MI455X (gfx1250) — compile-verified
//
#include <hip/hip_runtime.h>
#include <stdint.h>

// Simplex projection along last axis:  out = VALUE * proj_simplex(x / VALUE)
//   = relu(x - VALUE * tau)   where tau solves  sum_i relu(x_i/VALUE - tau) = 1
//
// One wave32 per row of 1024 fp32. Rows staged HBM -> LDS via gfx1250 async
// loads (ASYNCcnt path), then register-resident Newton iteration with wave32
// shuffle reductions.

#define N_COLS         1024
#define ROWS_PER_BLOCK 8     // 8 waves * 32 lanes = 256 threads
#define VALUE_F        0.1f
#define INV_VALUE      10.0f

// gfx1250 async global->LDS copy, 16B per lane.
// VDST operand = wave-relative LDS byte address (low 32 bits of the generic
// pointer into the LDS aperture, per CDNA5 ISA 10.2). INST_OFFSET is applied
// to BOTH the LDS and global addresses (ISA 08 async pseudocode), so all 8
// chunks share one address pair and differ only in the immediate.
template <int BYTE_OFF>
__device__ __forceinline__ void async_load_b128(const float* gaddr, const float* lds_generic) {
    unsigned lds_off = (unsigned)(size_t)lds_generic;
    asm volatile("global_load_async_to_lds_b128 %0, %1, off offset:%c2"
                 :: "v"(lds_off), "v"(gaddr), "i"(BYTE_OFF)
                 : "memory");
}

__device__ __forceinline__ void wait_asynccnt0() {
    asm volatile("s_wait_asynccnt 0x0" ::: "memory");
}

__global__ __launch_bounds__(ROWS_PER_BLOCK * 32, 8)
void simplex_proj_kernel(const float* __restrict__ x,
                         float* __restrict__ out,
                         int nrows) {
    __shared__ float lds[ROWS_PER_BLOCK * N_COLS];   // 32 KB

    const int tid  = threadIdx.x;
    const int w    = tid >> 5;          // wave id within block (row slot)
    const int lane = tid & 31;
    const int row  = blockIdx.x * ROWS_PER_BLOCK + w;
    if (row >= nrows) return;           // wave-uniform

    const float* grow = x + (size_t)row * N_COLS;
    float*       lrow = &lds[w * N_COLS];

    // ---- stage this wave's row into its private LDS slice (8 x b128/lane) ----
    // lane handles bytes [lane*16 + k*512], k = 0..7, on both sides.
    {
        const float* gbase = grow + lane * 4;
        const float* lbase = lrow + lane * 4;
        async_load_b128<0 * 512>(gbase, lbase);
        async_load_b128<1 * 512>(gbase, lbase);
        async_load_b128<2 * 512>(gbase, lbase);
        async_load_b128<3 * 512>(gbase, lbase);
        async_load_b128<4 * 512>(gbase, lbase);
        async_load_b128<5 * 512>(gbase, lbase);
        async_load_b128<6 * 512>(gbase, lbase);
        async_load_b128<7 * 512>(gbase, lbase);
    }
    wait_asynccnt0();

    // ---- pull the row into registers, scale by 1/VALUE, reduce sum ----
    float z[32];
    float s = 0.0f;
    {
        const float4* l4 = (const float4*)lrow;
#pragma unroll
        for (int k = 0; k < 8; ++k) {
            float4 v = l4[lane + k * 32];            // ds_load_b128
            z[k * 4 + 0] = v.x * INV_VALUE;
            z[k * 4 + 1] = v.y * INV_VALUE;
            z[k * 4 + 2] = v.z * INV_VALUE;
            z[k * 4 + 3] = v.w * INV_VALUE;
#pragma unroll
            for (int j = 0; j < 4; ++j)
                s += z[k * 4 + j];
        }
    }
    // wave32 butterfly reduction (all lanes end with full-row sum)
#pragma unroll
    for (int off = 16; off > 0; off >>= 1)
        s += __shfl_xor(s, off, 32);

    // ---- Newton on g(tau) = sum relu(z - tau) - 1 (convex, piecewise linear).
    // tau0 = (sum-1)/n <= tau*; iterates increase monotonically to exact root.
    float tau = (s - 1.0f) * (1.0f / (float)N_COLS);
#pragma unroll 1
    for (int it = 0; it < 64; ++it) {
        float g = 0.0f;
        int   c = 0;
#pragma unroll
        for (int i = 0; i < 32; ++i) {
            float d = z[i] - tau;
            g += fmaxf(d, 0.0f);
            c += (d > 0.0f) ? 1 : 0;
        }
#pragma unroll
        for (int off = 16; off > 0; off >>= 1) {
            g += __shfl_xor(g, off, 32);
            c += __shfl_xor(c, off, 32);
        }
        if (c == 0) break;                       // fp safety; tau* < max guarantees c>=1
        float tn = tau + (g - 1.0f) / (float)c;
        if (!(tn > tau)) break;                  // converged (wave-uniform branch)
        tau = tn;
    }

    // ---- write result: out = VALUE * relu(z - tau), coalesced b128 stores ----
    float4* o4 = (float4*)(out + (size_t)row * N_COLS);
#pragma unroll
    for (int k = 0; k < 8; ++k) {
        float4 o;
        o.x = fmaxf(z[k * 4 + 0] - tau, 0.0f) * VALUE_F;
        o.y = fmaxf(z[k * 4 + 1] - tau, 0.0f) * VALUE_F;
        o.z = fmaxf(z[k * 4 + 2] - tau, 0.0f) * VALUE_F;
        o.w = fmaxf(z[k * 4 + 3] - tau, 0.0f) * VALUE_F;
        o4[lane + k * 32] = o;
    }
}

extern "C" void kernel_launch(void* const* d_in, const int* in_sizes, int n_in,
                              void* d_out, int out_size, void* d_ws, size_t ws_size,
                              hipStream_t stream) {
    const float* x = (const float*)d_in[0];
    float* out = (float*)d_out;
    const int nrows = in_sizes[0] / N_COLS;      // 32*2048 = 65536
    const int blocks = (nrows + ROWS_PER_BLOCK - 1) / ROWS_PER_BLOCK;
    simplex_proj_kernel<<<blocks, ROWS_PER_BLOCK * 32, 0, stream>>>(x, out, nrows);
}